// Contracter_33818572489309
// MI455X (gfx1250) — compile-verified
//
#include <hip/hip_runtime.h>

// Problem constants (from the reference module configuration)
#define BASE     9            // base irrep dim (0e+1o+2e = 1+3+5)
#define MULCH    32           // multiplicity (channels u)
#define FDIM     (MULCH*BASE) // 288 features per edge
#define NPATH    11           // tensor-product paths
#define NNODES   10000        // scatter_dim_size (fixed in reference)
#define KPAD     84           // K = 81 (i*9+j) padded to 21 steps of 4
#define NSTEP    21
#define NPADN    16           // N = 9 padded to 16

typedef float v2f __attribute__((ext_vector_type(2)));
typedef float v8f __attribute__((ext_vector_type(8)));

// ---------------- zero the node accumulator ----------------
__global__ void zero_kernel(float* __restrict__ p, int n) {
    int i = blockIdx.x * blockDim.x + threadIdx.x;
    if (i < n) p[i] = 0.0f;
}

// ---------------- segment_sum: scatter x2 into nodes (fp32 HW atomics) ----------------
__global__ void scatter_kernel(const float* __restrict__ x2,
                               const int*   __restrict__ idxs,
                               float*       __restrict__ nodes,
                               int nElems) {
    int i = blockIdx.x * blockDim.x + threadIdx.x;
    if (i < nElems) {
        int e = i / FDIM;
        int f = i - e * FDIM;
        unsafeAtomicAdd(&nodes[(size_t)idxs[e] * FDIM + f], x2[i]);
    }
}

// ---------------- ww3j: combine per-channel path weights with w3j, pad to B layout ----
// wpad[u][kk][n], kk = i*9 + j in [0,84), n in [0,16); zero outside (81, 9).
__global__ void wpad_kernel(const float* __restrict__ w3j,
                            const float* __restrict__ weights,
                            float*       __restrict__ wpad) {
    int u = blockIdx.x;
    for (int s = threadIdx.x; s < KPAD * NPADN; s += blockDim.x) {
        int kk = s / NPADN;
        int n  = s - kk * NPADN;
        float acc = 0.0f;
        if (kk < BASE * BASE && n < BASE) {
            int i = kk / BASE;
            int j = kk - i * BASE;
#pragma unroll
            for (int p = 0; p < NPATH; ++p)
                acc += weights[u * NPATH + p] *
                       w3j[((p * BASE + i) * BASE + j) * BASE + n];
        }
        wpad[u * KPAD * NPADN + s] = acc;
    }
}

// ---------------- fused gather + bilinear contraction via V_WMMA_F32_16X16X4_F32 ------
// Per 16-edge tile and channel u:  D[16e x 16n] = sum_t A_t[16e x 4k] * B_t[4k x 16n]
// A_t: outer products x1[e,u,i]*x2g[e,u,j] with k = i*9+j; B_t: wpad[u] slices.
__global__ void __launch_bounds__(256)
contract_kernel(const float* __restrict__ x1,
                const float* __restrict__ nodes,
                const int*   __restrict__ idxs,
                const float* __restrict__ wpad,
                float*       __restrict__ out,
                int E) {
    __shared__ float x1s[16 * FDIM];   // 16 edges x 288 features
    __shared__ float x2s[16 * FDIM];   // gathered node features per edge

    const int tid = threadIdx.x;
    const int e0  = blockIdx.x * 16;

    // Stage x1 tile (coalesced) and gathered x2 tile into LDS.
    for (int s = tid; s < 16 * FDIM; s += 256) {
        int m = s / FDIM;
        int f = s - m * FDIM;
        int e = e0 + m;
        float a = 0.0f, b = 0.0f;
        if (e < E) {
            a = x1[(size_t)e * FDIM + f];
            b = nodes[(size_t)idxs[e] * FDIM + f];
        }
        x1s[s] = a;
        x2s[s] = b;
    }
    __syncthreads();

    const int lane = tid & 31;
    const int wave = tid >> 5;     // 8 waves per block
    const int half = lane >> 4;    // lane group: 0 -> K pair {4t,4t+1}, 1 -> {4t+2,4t+3}
    const int mn   = lane & 15;    // M index for A, N index for B/D

    // Each wave handles 4 consecutive channels.
#pragma unroll 1
    for (int uu = 0; uu < 4; ++uu) {
        const int u = wave * 4 + uu;
        const float* xs1 = &x1s[mn * FDIM + u * BASE];
        const float* xs2 = &x2s[mn * FDIM + u * BASE];
        const float* wb  = &wpad[u * KPAD * NPADN + mn];

        v8f acc = {};
#pragma unroll
        for (int t = 0; t < NSTEP; ++t) {
            const int k0 = 4 * t + 2 * half;
            // A fragment: per-lane K = k0, k0+1 (ISA 16x4 A layout)
            float a0 = 0.0f, a1 = 0.0f;
            if (k0 < BASE * BASE) {
                int i = k0 / BASE, j = k0 - i * BASE;
                a0 = xs1[i] * xs2[j];
            }
            if (k0 + 1 < BASE * BASE) {
                int i = (k0 + 1) / BASE, j = (k0 + 1) - i * BASE;
                a1 = xs1[i] * xs2[j];
            }
            v2f a; a.x = a0; a.y = a1;
            // B fragment: rows K = k0, k0+1, column mn (zero-padded buffer)
            v2f b;
            b.x = wb[(k0 + 0) * NPADN];
            b.y = wb[(k0 + 1) * NPADN];

            acc = __builtin_amdgcn_wmma_f32_16x16x4_f32(
                /*neg_a=*/false, a, /*neg_b=*/false, b,
                /*c_mod=*/(short)0, acc, /*reuse_a=*/false, /*reuse_b=*/false);
        }

        // D layout: VGPR r -> M = r + 8*half, N = mn (valid N < 9)
        if (mn < BASE) {
#pragma unroll
            for (int r = 0; r < 8; ++r) {
                int e = e0 + r + 8 * half;
                if (e < E) out[(size_t)e * FDIM + u * BASE + mn] = acc[r];
            }
        }
    }
}

extern "C" void kernel_launch(void* const* d_in, const int* in_sizes, int n_in,
                              void* d_out, int out_size, void* d_ws, size_t ws_size,
                              hipStream_t stream) {
    const float* x1      = (const float*)d_in[0];
    const float* x2      = (const float*)d_in[1];
    const int*   idxs    = (const int*)d_in[2];
    // d_in[3] = scatter_dim_size (device scalar); fixed at 10000 in the reference.
    const float* w3j     = (const float*)d_in[4];
    const float* weights = (const float*)d_in[5];
    float*       out     = (float*)d_out;

    const int E = in_sizes[0] / FDIM;

    // Workspace layout: [node accumulator | padded weight matrices]
    float* nodes = (float*)d_ws;
    float* wpad  = nodes + (size_t)NNODES * FDIM;

    const int nodeElems = NNODES * FDIM;
    zero_kernel<<<(nodeElems + 255) / 256, 256, 0, stream>>>(nodes, nodeElems);

    const int sElems = E * FDIM;
    scatter_kernel<<<(sElems + 255) / 256, 256, 0, stream>>>(x2, idxs, nodes, sElems);

    wpad_kernel<<<MULCH, 256, 0, stream>>>(w3j, weights, wpad);

    const int tiles = (E + 15) / 16;
    contract_kernel<<<tiles, 256, 0, stream>>>(x1, nodes, idxs, wpad, out, E);
}